// LowrankVideo_62740882260199
// MI455X (gfx1250) — compile-verified
//
#include <hip/hip_runtime.h>

namespace {

typedef unsigned u32x4 __attribute__((ext_vector_type(4)));
typedef unsigned u32x8 __attribute__((ext_vector_type(8)));

constexpr int kN     = 262144;
constexpr int kOD    = 2;
constexpr int kTR    = 8;
constexpr int kRANK  = 16;
constexpr int kRES   = 256;
constexpr int kTRES  = 300;
constexpr int kFDIM  = 32;
constexpr int kFRES  = 512;
constexpr int kC     = kOD * kTR * kRANK;          // 256 channels
constexpr int kTimeF = kOD * kTR * kTRES;          // 4800 floats (19.2 KB)
constexpr int kTimeHalf = kTimeF / 2;              // 2400 floats: TDM half
constexpr int kBins  = 4096;                       // 4 bits per axis
constexpr long long kGridElems = (long long)kC * kRES * kRES;      // 16,777,216
constexpr long long kFeatElems = (long long)kFDIM * kFRES * kFRES; // 8,388,608

// align_corners=True, border padding: u in [0, R-1]
__device__ __forceinline__ void axis01(float v, int R, int& i0, int& i1, float& w) {
  float u = (v + 1.0f) * 0.5f * (float)(R - 1);
  u = fminf(fmaxf(u, 0.0f), (float)(R - 1));
  float f = floorf(u);
  i0 = (int)f;
  i1 = min(i0 + 1, R - 1);
  w = u - f;
}

__device__ __forceinline__ float4 bl4(float4 a, float4 b, float4 c, float4 d,
                                      float wa, float wb, float wc, float wd) {
  float4 r;
  r.x = wa * a.x + wb * b.x + wc * c.x + wd * d.x;
  r.y = wa * a.y + wb * b.y + wc * c.y + wd * d.y;
  r.z = wa * a.z + wb * b.z + wc * c.z + wd * d.z;
  r.w = wa * a.w + wb * b.w + wc * c.w + wd * d.w;
  return r;
}

struct Bil {
  int o00, o01, o10, o11;
  float w00, w01, w10, w11;
};

__device__ __forceinline__ Bil mk_bil(int x0, int x1, float wx,
                                      int y0, int y1, float wy,
                                      int rowlen, int vstride) {
  Bil b;
  float iwx = 1.0f - wx, iwy = 1.0f - wy;
  b.w00 = iwy * iwx; b.w01 = iwy * wx; b.w10 = wy * iwx; b.w11 = wy * wx;
  b.o00 = (y0 * rowlen + x0) * vstride;
  b.o01 = (y0 * rowlen + x1) * vstride;
  b.o10 = (y1 * rowlen + x0) * vstride;
  b.o11 = (y1 * rowlen + x1) * vstride;
  return b;
}

// Coarse 3D cell id: 4 bits per axis -> 4096 bins. Points inside one bin touch
// a ~900 KB slab of the three planes -> near-perfect L2 residency in the sampler.
__device__ __forceinline__ int bin_of(const float* __restrict__ pts, int p) {
  float px = pts[3 * p + 0], py = pts[3 * p + 1], pz = pts[3 * p + 2];
  int a0, a1, b0, b1, c0, c1; float w;
  axis01(px, kRES, a0, a1, w);
  axis01(py, kRES, b0, b1, w);
  axis01(pz, kRES, c0, c1, w);
  return ((a0 >> 4) << 8) | ((b0 >> 4) << 4) | (c0 >> 4);
}

// ---------------- layout transpose: [C, HW] -> [HW, C] -------------------
__global__ __launch_bounds__(256)
void lrv_transpose(const float* __restrict__ in, float* __restrict__ outp,
                   int Cdim, int HW) {
  __shared__ float tile[32][33];
  const int hw0 = blockIdx.x * 32;
  const int c0  = blockIdx.y * 32;
  const int tx = threadIdx.x, ty = threadIdx.y;   // ty in [0,8)
#pragma unroll
  for (int j = 0; j < 4; ++j)
    tile[ty + 8 * j][tx] =
        __builtin_nontemporal_load(&in[(size_t)(c0 + ty + 8 * j) * HW + hw0 + tx]);
  __syncthreads();
#pragma unroll
  for (int j = 0; j < 4; ++j)
    outp[(size_t)(hw0 + ty + 8 * j) * Cdim + c0 + tx] = tile[tx][ty + 8 * j];
}

// ---------------- point binning (counting sort) --------------------------
__global__ __launch_bounds__(256)
void lrv_zero(unsigned* __restrict__ h) {
  h[blockIdx.x * 256 + threadIdx.x] = 0u;
}

__global__ __launch_bounds__(256)
void lrv_hist(const float* __restrict__ pts, unsigned* __restrict__ hist) {
  const int p = blockIdx.x * 256 + threadIdx.x;
  atomicAdd(&hist[bin_of(pts, p)], 1u);
}

// One block, 1024 threads: exclusive scan of 4096 bin counts.
__global__ __launch_bounds__(1024)
void lrv_scan(const unsigned* __restrict__ hist, unsigned* __restrict__ offs) {
  __shared__ unsigned buf[1024];
  const int t = threadIdx.x;
  unsigned v[4];
#pragma unroll
  for (int j = 0; j < 4; ++j) v[j] = hist[4 * t + j];
  buf[t] = v[0] + v[1] + v[2] + v[3];
  __syncthreads();
  for (int d = 1; d < 1024; d <<= 1) {       // Hillis-Steele inclusive scan
    unsigned x = (t >= d) ? buf[t - d] : 0u;
    __syncthreads();
    buf[t] += x;
    __syncthreads();
  }
  unsigned run = (t == 0) ? 0u : buf[t - 1]; // exclusive
#pragma unroll
  for (int j = 0; j < 4; ++j) { offs[4 * t + j] = run; run += v[j]; }
}

__global__ __launch_bounds__(256)
void lrv_scatter(const float* __restrict__ pts, unsigned* __restrict__ offs,
                 unsigned* __restrict__ perm) {
  const int p = blockIdx.x * 256 + threadIdx.x;
  const unsigned pos = atomicAdd(&offs[bin_of(pts, p)], 1u);
  perm[pos] = (unsigned)p;
}

// ---------------- fused sampler ------------------------------------------
// TL = true: tables are [H, W, C] in scratch; TL = false: original [C, H, W].
template <bool TL>
__global__ __launch_bounds__(256)
void lrv_sample(const float* __restrict__ pts,
                const float* __restrict__ tstamps,
                const float* __restrict__ g0,
                const float* __restrict__ g1,
                const float* __restrict__ g2,
                const float* __restrict__ tc,
                const float* __restrict__ feat,
                const unsigned* __restrict__ perm,
                float* __restrict__ out) {
  __shared__ float ltime[kTimeF];

  const int tid = threadIdx.x;

  // ---- Stage time_coef (19.2 KB) into LDS with BOTH CDNA5 async engines.
  // Half 1 (2400 f32): one Tensor Data Mover DMA, issued by wave 0 off SGPR
  //   descriptors (TENSORcnt). D# group0: count=1, lds_addr, global_addr,
  //   type=2. D# group1: data_size=4B, tensor_dim0=tile_dim0=2400,
  //   tensor_dim1=tile_dim1=1, strides=2400. 2-group form => tensor up to 2D.
  // Half 2 (2400 f32): per-lane global_load_async_to_lds_b128 (ASYNCcnt).
  // Both waits are deferred until after the channel loop.
  if (tid < 32) {
    unsigned long long ga = (unsigned long long)tc;
    unsigned ldsb = (unsigned)(size_t)(void*)ltime;       // low 32b = LDS byte addr
    u32x4 d0 = { 1u,                                      // count=1, user mode
                 ldsb,                                    // lds_addr
                 (unsigned)ga,                            // global_addr[31:0]
                 ((unsigned)(ga >> 32) & 0x01FFFFFFu) | 0x80000000u }; // addr[56:32] | type=2
    u32x8 d1 = { 0x00020000u,                             // data_size=2 (4 bytes)
                 (unsigned)kTimeHalf << 16,               // tensor_dim0[15:0]=2400
                 0x00010000u,                             // tensor_dim1=1
                 (unsigned)kTimeHalf << 16,               // tile_dim0=2400
                 0x00000001u,                             // tile_dim1=1
                 (unsigned)kTimeHalf,                     // tensor_dim0_stride=2400
                 (unsigned)kTimeHalf << 16,               // tensor_dim1_stride=2400
                 0u };
    asm volatile("tensor_load_to_lds %0, %1" :: "s"(d0), "s"(d1) : "memory");
  }
  {
    const float4* src = (const float4*)tc;
    float4* dst = (float4*)ltime;
    for (int i = kTimeHalf / 4 + tid; i < kTimeF / 4; i += 256) {
      unsigned ldsoff = (unsigned)(size_t)(const void*)(dst + i);
      asm volatile("global_load_async_to_lds_b128 %0, %1, off"
                   :: "v"(ldsoff), "v"(src + i)
                   : "memory");
    }
  }

  const int gid = blockIdx.x * 256 + tid;
  const int p = perm ? (int)perm[gid] : gid;    // locality-sorted point id
  const float px = pts[3 * p + 0];
  const float py = pts[3 * p + 1];
  const float pz = pts[3 * p + 2];
  const float ts = tstamps[p];

  int a0, a1, b0, b1, c0, c1;
  float wa, wb, wc;
  axis01(px, kRES, a0, a1, wa);
  axis01(py, kRES, b0, b1, wb);
  axis01(pz, kRES, c0, c1, wc);

  // coo_combs: grid0 <- (x=pts0, y=pts1); grid1 <- (x=pts0, y=pts2); grid2 <- (x=pts1, y=pts2)
  const int vs = TL ? kC : 1;
  Bil B0 = mk_bil(a0, a1, wa, b0, b1, wb, kRES, vs);
  Bil B1 = mk_bil(a0, a1, wa, c0, c1, wc, kRES, vs);
  Bil B2 = mk_bil(b0, b1, wb, c0, c1, wc, kRES, vs);

  float racc[kOD * kTR];
#pragma unroll
  for (int i = 0; i < kOD * kTR; ++i) racc[i] = 0.0f;

  if (TL) {
    const float4* q0a = (const float4*)(g0 + B0.o00);
    const float4* q0b = (const float4*)(g0 + B0.o01);
    const float4* q0c = (const float4*)(g0 + B0.o10);
    const float4* q0d = (const float4*)(g0 + B0.o11);
    const float4* q1a = (const float4*)(g1 + B1.o00);
    const float4* q1b = (const float4*)(g1 + B1.o01);
    const float4* q1c = (const float4*)(g1 + B1.o10);
    const float4* q1d = (const float4*)(g1 + B1.o11);
    const float4* q2a = (const float4*)(g2 + B2.o00);
    const float4* q2b = (const float4*)(g2 + B2.o01);
    const float4* q2c = (const float4*)(g2 + B2.o10);
    const float4* q2d = (const float4*)(g2 + B2.o11);
#pragma unroll 2
    for (int ch = 0; ch < kC / 4; ++ch) {
      float4 s0 = bl4(q0a[ch], q0b[ch], q0c[ch], q0d[ch], B0.w00, B0.w01, B0.w10, B0.w11);
      float4 s1 = bl4(q1a[ch], q1b[ch], q1c[ch], q1d[ch], B1.w00, B1.w01, B1.w10, B1.w11);
      float4 s2 = bl4(q2a[ch], q2b[ch], q2c[ch], q2d[ch], B2.w00, B2.w01, B2.w10, B2.w11);
      float4 pr;
      pr.x = s0.x * s1.x * s2.x;
      pr.y = s0.y * s1.y * s2.y;
      pr.z = s0.z * s1.z * s2.z;
      pr.w = s0.w * s1.w * s2.w;
      racc[ch >> 2] += (pr.x + pr.y) + (pr.z + pr.w);   // rank-16 partial sum
    }
  } else {
#pragma unroll 4
    for (int c = 0; c < kC; ++c) {
      const int off = c * (kRES * kRES);
      float s0 = B0.w00 * g0[off + B0.o00] + B0.w01 * g0[off + B0.o01] +
                 B0.w10 * g0[off + B0.o10] + B0.w11 * g0[off + B0.o11];
      float s1 = B1.w00 * g1[off + B1.o00] + B1.w01 * g1[off + B1.o01] +
                 B1.w10 * g1[off + B1.o10] + B1.w11 * g1[off + B1.o11];
      float s2 = B2.w00 * g2[off + B2.o00] + B2.w01 * g2[off + B2.o01] +
                 B2.w10 * g2[off + B2.o10] + B2.w11 * g2[off + B2.o11];
      racc[c >> 4] += s0 * s1 * s2;
    }
  }

  // ltime is needed now: issuing wave drains TENSORcnt, every wave drains its
  // own ASYNCcnt, then the workgroup barrier publishes LDS.
  if (tid < 32) {
    asm volatile("s_wait_tensorcnt 0x0" ::: "memory");
  }
  asm volatile("s_wait_asynccnt 0x0" ::: "memory");
  __syncthreads();

  int t0, t1;
  float wt;
  axis01(ts, kTRES, t0, t1, wt);
  const float iwt = 1.0f - wt;

  float interp[kOD];
#pragma unroll
  for (int o = 0; o < kOD; ++o) {
    float acc = 0.0f;
#pragma unroll
    for (int t = 0; t < kTR; ++t) {
      const int k = o * kTR + t;
      const float tv = ltime[k * kTRES + t0] * iwt + ltime[k * kTRES + t1] * wt;
      acc += racc[k] * tv;
    }
    interp[o] = acc;
  }

  // final feature bilerp at (x=interp[0], y=interp[1])
  int fx0, fx1, fy0, fy1;
  float wfx, wfy;
  axis01(interp[0], kFRES, fx0, fx1, wfx);
  axis01(interp[1], kFRES, fy0, fy1, wfy);
  Bil F = mk_bil(fx0, fx1, wfx, fy0, fy1, wfy, kFRES, TL ? kFDIM : 1);

  if (TL) {
    const float4* f00 = (const float4*)(feat + F.o00);
    const float4* f01 = (const float4*)(feat + F.o01);
    const float4* f10 = (const float4*)(feat + F.o10);
    const float4* f11 = (const float4*)(feat + F.o11);
    float4* op = (float4*)(out + (size_t)p * kFDIM);
#pragma unroll
    for (int j = 0; j < kFDIM / 4; ++j)
      op[j] = bl4(f00[j], f01[j], f10[j], f11[j], F.w00, F.w01, F.w10, F.w11);
  } else {
#pragma unroll 4
    for (int f = 0; f < kFDIM; ++f) {
      const int off = f * (kFRES * kFRES);
      out[(size_t)p * kFDIM + f] =
          F.w00 * feat[off + F.o00] + F.w01 * feat[off + F.o01] +
          F.w10 * feat[off + F.o10] + F.w11 * feat[off + F.o11];
    }
  }
}

}  // namespace

extern "C" void kernel_launch(void* const* d_in, const int* in_sizes, int n_in,
                              void* d_out, int out_size, void* d_ws, size_t ws_size,
                              hipStream_t stream) {
  (void)in_sizes; (void)n_in; (void)out_size;
  const float* pts  = (const float*)d_in[0];   // [N, 3]
  const float* ts   = (const float*)d_in[1];   // [N]
  const float* g0   = (const float*)d_in[2];   // [256, 256, 256]
  const float* g1   = (const float*)d_in[3];
  const float* g2   = (const float*)d_in[4];
  const float* tc   = (const float*)d_in[5];   // [16, 300]
  const float* feat = (const float*)d_in[6];   // [32, 512, 512]
  float* out = (float*)d_out;                  // [N, 32]

  const size_t tablesF = (size_t)(3 * kGridElems + kFeatElems);          // floats
  const size_t binF    = (size_t)kN + 2 * kBins;                         // perm+hist+offs
  const size_t needFull = (tablesF + binF) * sizeof(float);              // ~225 MiB
  const size_t needBin  = binF * sizeof(float);                          // ~1.03 MiB

  const float* sg0 = g0; const float* sg1 = g1; const float* sg2 = g2;
  const float* sfe = feat;
  unsigned* perm = nullptr; unsigned* hist = nullptr; unsigned* offs = nullptr;
  bool tl = false;

  if (ws_size >= needFull) {
    float* g0T = (float*)d_ws;
    float* g1T = g0T + kGridElems;
    float* g2T = g1T + kGridElems;
    float* fT  = g2T + kGridElems;
    perm = (unsigned*)(fT + kFeatElems);
    hist = perm + kN;
    offs = hist + kBins;
    dim3 tb(32, 8);
    lrv_transpose<<<dim3(kRES * kRES / 32, kC / 32), tb, 0, stream>>>(g0, g0T, kC, kRES * kRES);
    lrv_transpose<<<dim3(kRES * kRES / 32, kC / 32), tb, 0, stream>>>(g1, g1T, kC, kRES * kRES);
    lrv_transpose<<<dim3(kRES * kRES / 32, kC / 32), tb, 0, stream>>>(g2, g2T, kC, kRES * kRES);
    lrv_transpose<<<dim3(kFRES * kFRES / 32, kFDIM / 32), tb, 0, stream>>>(feat, fT, kFDIM, kFRES * kFRES);
    sg0 = g0T; sg1 = g1T; sg2 = g2T; sfe = fT;
    tl = true;
  } else if (ws_size >= needBin) {
    perm = (unsigned*)d_ws;
    hist = perm + kN;
    offs = hist + kBins;
  }

  if (perm) {
    lrv_zero<<<kBins / 256, 256, 0, stream>>>(hist);
    lrv_hist<<<kN / 256, 256, 0, stream>>>(pts, hist);
    lrv_scan<<<1, 1024, 0, stream>>>(hist, offs);
    lrv_scatter<<<kN / 256, 256, 0, stream>>>(pts, offs, perm);
  }

  if (tl) {
    lrv_sample<true><<<kN / 256, 256, 0, stream>>>(pts, ts, sg0, sg1, sg2, tc, sfe, perm, out);
  } else {
    lrv_sample<false><<<kN / 256, 256, 0, stream>>>(pts, ts, sg0, sg1, sg2, tc, sfe, perm, out);
  }
}